// GCN_13365938226013
// MI455X (gfx1250) — compile-verified
//
#include <hip/hip_runtime.h>
#include <hip/hip_bf16.h>

typedef __attribute__((ext_vector_type(2))) float v2f;
typedef __attribute__((ext_vector_type(8))) float v8f;

#define F 128

// ---------------- degree / normalization ----------------

__global__ void __launch_bounds__(256) gcn_deg_init(float* __restrict__ deg, int n) {
  int i = blockIdx.x * blockDim.x + threadIdx.x;
  if (i < n) deg[i] = 1.0f;  // self-loop contribution
}

__global__ void __launch_bounds__(256) gcn_deg_accum(const long long* __restrict__ dst,
                                                     float* __restrict__ deg, int E) {
  int e = blockIdx.x * blockDim.x + threadIdx.x;
  if (e < E) atomicAdd(&deg[(int)dst[e]], 1.0f);
}

__global__ void __launch_bounds__(256) gcn_deg_fin(float* __restrict__ deg, int n) {
  int i = blockIdx.x * blockDim.x + threadIdx.x;
  if (i < n) deg[i] = rsqrtf(deg[i]);  // deg >= 1 always (self-loops)
}

// ---------------- fp32 WMMA GEMM: H = X @ W, X:[n,128], W:[128,128] ----------------
// Block: 256 threads = 8 waves. Block tile: 64 rows x 128 cols.
// Wave w owns N-tile w (cols 16w..16w+15) and all 4 M-tiles (rows 0..63 of block).

__global__ void __launch_bounds__(256) gcn_gemm_wmma(const float* __restrict__ X,
                                                     const float* __restrict__ W,
                                                     float* __restrict__ H, int n) {
  __shared__ float As[64][36];   // 64 x 32 k-chunk, +4 pad (stride 36 -> conflict-free col reads)
  __shared__ float Bs[32][F];    // k-chunk x 128 cols (row reads are contiguous, no pad needed)

  const int tid  = threadIdx.x;
  const int wave = tid >> 5;
  const int lane = tid & 31;
  const int half = lane >> 4;    // 0: K pair {0,1}; 1: K pair {2,3}
  const int l16  = lane & 15;
  const int row0 = blockIdx.x * 64;

  v8f acc0 = {}, acc1 = {}, acc2 = {}, acc3 = {};

  // cooperative-load coordinates
  const int ar = tid >> 2;           // 0..63  (A row in tile)
  const int ac = (tid & 3) * 8;      // 0,8,16,24 (A col base, 8 floats per thread)
  const int bk = tid >> 3;           // 0..31  (B row in chunk)
  const int bn = (tid & 7) * 16;     // col base, 16 floats per thread

  for (int kc = 0; kc < F; kc += 32) {
    // ---- stage A tile ----
    const int grow = row0 + ar;
    if (grow < n) {
      const float4* s = (const float4*)(X + (size_t)grow * F + kc + ac);
      float4 p0 = s[0], p1 = s[1];
      As[ar][ac + 0] = p0.x; As[ar][ac + 1] = p0.y; As[ar][ac + 2] = p0.z; As[ar][ac + 3] = p0.w;
      As[ar][ac + 4] = p1.x; As[ar][ac + 5] = p1.y; As[ar][ac + 6] = p1.z; As[ar][ac + 7] = p1.w;
    } else {
      #pragma unroll
      for (int j = 0; j < 8; ++j) As[ar][ac + j] = 0.0f;
    }
    // ---- stage B (W) chunk ----
    {
      const float4* s = (const float4*)(W + (size_t)(kc + bk) * F + bn);
      float4 q0 = s[0], q1 = s[1], q2 = s[2], q3 = s[3];
      float4* d = (float4*)&Bs[bk][bn];
      d[0] = q0; d[1] = q1; d[2] = q2; d[3] = q3;
    }
    __syncthreads();

    #pragma unroll
    for (int kk = 0; kk < 32; kk += 4) {
      const int k2 = kk + 2 * half;
      v2f b;
      b[0] = Bs[k2 + 0][wave * 16 + l16];
      b[1] = Bs[k2 + 1][wave * 16 + l16];
      v2f a0; a0[0] = As[ 0 + l16][k2]; a0[1] = As[ 0 + l16][k2 + 1];
      v2f a1; a1[0] = As[16 + l16][k2]; a1[1] = As[16 + l16][k2 + 1];
      v2f a2; a2[0] = As[32 + l16][k2]; a2[1] = As[32 + l16][k2 + 1];
      v2f a3; a3[0] = As[48 + l16][k2]; a3[1] = As[48 + l16][k2 + 1];
      acc0 = __builtin_amdgcn_wmma_f32_16x16x4_f32(false, a0, false, b, (short)0, acc0, false, false);
      acc1 = __builtin_amdgcn_wmma_f32_16x16x4_f32(false, a1, false, b, (short)0, acc1, false, false);
      acc2 = __builtin_amdgcn_wmma_f32_16x16x4_f32(false, a2, false, b, (short)0, acc2, false, false);
      acc3 = __builtin_amdgcn_wmma_f32_16x16x4_f32(false, a3, false, b, (short)0, acc3, false, false);
    }
    __syncthreads();
  }

  // ---- store D tiles: VGPR r -> M = r (lanes 0-15) / M = 8+r (lanes 16-31), N = l16 ----
  const int col = wave * 16 + l16;
  #pragma unroll
  for (int r = 0; r < 8; ++r) {
    const int rr = r + 8 * half;
    const int row = row0 + rr;
    if (row      < n) H[(size_t)(row     ) * F + col] = acc0[r];
    if (row + 16 < n) H[(size_t)(row + 16) * F + col] = acc1[r];
    if (row + 32 < n) H[(size_t)(row + 32) * F + col] = acc2[r];
    if (row + 48 < n) H[(size_t)(row + 48) * F + col] = acc3[r];
  }
}

// ---------------- aggregation ----------------

// out[i,f] = bias[f] + H[i,f] * dis[i]^2   (self-loop term; also fully initializes out)
__global__ void __launch_bounds__(256) gcn_self_bias(const float* __restrict__ H,
                                                     const float* __restrict__ bias,
                                                     const float* __restrict__ dis,
                                                     float* __restrict__ out, int n) {
  int i = blockIdx.x * blockDim.x + threadIdx.x;
  if (i < n * F) {
    int node = i >> 7, f = i & (F - 1);
    float d = dis[node];
    out[i] = fmaf(H[i], d * d, bias[f]);
  }
}

// one wave per edge; each lane handles 4 features (float4 gather + 4 fp32 atomics)
__global__ void __launch_bounds__(256) gcn_scatter(const float* __restrict__ H,
                                                   const long long* __restrict__ src,
                                                   const long long* __restrict__ dst,
                                                   const float* __restrict__ dis,
                                                   float* __restrict__ out, int E) {
  unsigned int tid = blockIdx.x * blockDim.x + threadIdx.x;
  unsigned int e = tid >> 5;
  int g = (tid & 31) * 4;
  if (e >= (unsigned int)E) return;
  int s = (int)src[e];
  int d = (int)dst[e];
  float norm = dis[s] * dis[d];
  float4 v = *(const float4*)(H + (size_t)s * F + g);
  float* o = out + (size_t)d * F + g;
  atomicAdd(o + 0, v.x * norm);
  atomicAdd(o + 1, v.y * norm);
  atomicAdd(o + 2, v.z * norm);
  atomicAdd(o + 3, v.w * norm);
}

__global__ void __launch_bounds__(256) gcn_relu(float* __restrict__ x, int total) {
  int i = blockIdx.x * blockDim.x + threadIdx.x;
  if (i < total) x[i] = fmaxf(x[i], 0.0f);
}

// ---------------- driver ----------------

extern "C" void kernel_launch(void* const* d_in, const int* in_sizes, int n_in,
                              void* d_out, int out_size, void* d_ws, size_t ws_size,
                              hipStream_t stream) {
  const float*     X  = (const float*)d_in[0];
  const long long* EI = (const long long*)d_in[1];
  const float*     W1 = (const float*)d_in[2];
  const float*     B1 = (const float*)d_in[3];
  const float*     W2 = (const float*)d_in[4];
  const float*     B2 = (const float*)d_in[5];
  float* OUT = (float*)d_out;

  const int n = in_sizes[0] / F;
  const int E = in_sizes[1] / 2;
  const long long* src = EI;
  const long long* dst = EI + E;

  // workspace layout: dis[n] | H[n*128] | A[n*128]   (~103 MB)
  float* dis  = (float*)d_ws;
  float* Hbuf = dis + (size_t)((n + 63) & ~63);
  float* Abuf = Hbuf + (size_t)n * F;

  const dim3 b256(256);
  const int gb_n    = (n + 255) / 256;
  const int gb_nf   = (n * F + 255) / 256;
  const int gb_e    = (E + 255) / 256;
  const unsigned int sc_threads = (unsigned int)E * 32u;
  const int gb_sc   = (int)((sc_threads + 255u) / 256u);
  const int gb_gemm = (n + 63) / 64;

  // degree / normalization
  gcn_deg_init <<<gb_n, b256, 0, stream>>>(dis, n);
  gcn_deg_accum<<<gb_e, b256, 0, stream>>>(dst, dis, E);
  gcn_deg_fin  <<<gb_n, b256, 0, stream>>>(dis, n);

  // layer 1
  gcn_gemm_wmma<<<gb_gemm, b256, 0, stream>>>(X, W1, Hbuf, n);
  gcn_self_bias<<<gb_nf, b256, 0, stream>>>(Hbuf, B1, dis, Abuf, n);
  gcn_scatter  <<<gb_sc, b256, 0, stream>>>(Hbuf, src, dst, dis, Abuf, E);
  gcn_relu     <<<gb_nf, b256, 0, stream>>>(Abuf, n * F);

  // layer 2
  gcn_gemm_wmma<<<gb_gemm, b256, 0, stream>>>(Abuf, W2, Hbuf, n);
  gcn_self_bias<<<gb_nf, b256, 0, stream>>>(Hbuf, B2, dis, OUT, n);
  gcn_scatter  <<<gb_sc, b256, 0, stream>>>(Hbuf, src, dst, dis, OUT, E);
  gcn_relu     <<<gb_nf, b256, 0, stream>>>(OUT, n * F);
}